// GeoNodeClassifier_32057635897949
// MI455X (gfx1250) — compile-verified
//
#include <hip/hip_runtime.h>
#include <hip/hip_bf16.h>

typedef __attribute__((ext_vector_type(2))) float v2f;
typedef __attribute__((ext_vector_type(8))) float v8f;

#define APAD 68    // 16 x 64 A tile, padded row stride
#define BPAD 260   // 64 x 256 B tile, padded row stride (breaks 2-way bank conflict)

// C[M x 256] = A[M x 64] * [root | W0 | W1 | W2]  (+ bias on cols 0..63)
// lda: row stride of A (64 for raw x, 256 when reading base cols of h_all in-place)
// relu_in: apply ReLU to A while staging (fuses layer-1 activation into layer-2 GEMM)
__global__ __launch_bounds__(256)
void rgcn_gemm_wmma(const float* __restrict__ A, int lda, int relu_in,
                    const float* __restrict__ root,   // [64][64]
                    const float* __restrict__ W,      // [3][64][64]
                    const float* __restrict__ bias,   // [64]
                    float* __restrict__ C)            // [M][256]
{
    __shared__ float As[16 * APAD];
    __shared__ float Bs[64 * BPAD];
    const int tid = threadIdx.x;
    const int m0  = blockIdx.x * 16;

    // Stage A tile (16 rows x 64 cols), coalesced
    for (int i = tid; i < 16 * 64; i += 256) {
        int r = i >> 6, c = i & 63;
        float v = A[(size_t)(m0 + r) * lda + c];
        if (relu_in) v = fmaxf(v, 0.0f);
        As[r * APAD + c] = v;
    }
    // Stage B tile (64 x 256): cols 0..63 <- root, cols 64+r*64+j <- W[r][k][j]
    for (int i = tid; i < 64 * 256; i += 256) {
        int k = i >> 8, j = i & 255;
        float v;
        if (j < 64) {
            v = root[k * 64 + j];
        } else {
            int jj = j - 64;
            v = W[(jj >> 6) * 4096 + k * 64 + (jj & 63)];
        }
        Bs[k * BPAD + j] = v;
    }
    __syncthreads();

    const int wave  = tid >> 5;
    const int lane  = tid & 31;
    const int lhalf = lane >> 4;   // 0: lanes 0-15, 1: lanes 16-31
    const int l16   = lane & 15;

    // Each wave produces two 16x16 output tiles: cols [wave*32, wave*32+32)
    const int col0 = wave * 32 + l16;
    const int col1 = col0 + 16;

    v8f acc0 = {};
    v8f acc1 = {};

    #pragma unroll
    for (int kk = 0; kk < 16; ++kk) {
        // A frag (16x4 f32): lanes 0-15 hold K=k0,k0+1; lanes 16-31 hold K=k0+2,k0+3
        const int kA = kk * 4 + lhalf * 2;
        v2f a, b0, b1;
        a.x  = As[l16 * APAD + kA];
        a.y  = As[l16 * APAD + kA + 1];
        // B frag (4x16 f32), symmetric K striping across lane halves
        b0.x = Bs[(kA    ) * BPAD + col0];
        b0.y = Bs[(kA + 1) * BPAD + col0];
        b1.x = Bs[(kA    ) * BPAD + col1];
        b1.y = Bs[(kA + 1) * BPAD + col1];
        acc0 = __builtin_amdgcn_wmma_f32_16x16x4_f32(false, a, false, b0,
                                                     (short)0, acc0, false, false);
        acc1 = __builtin_amdgcn_wmma_f32_16x16x4_f32(false, a, false, b1,
                                                     (short)0, acc1, false, false);
    }

    // C/D layout: VGPR v -> row v (lanes 0-15) or row 8+v (lanes 16-31)
    const int rbase = m0 + lhalf * 8;
    const float bv0 = (col0 < 64) ? bias[col0] : 0.0f;
    const float bv1 = (col1 < 64) ? bias[col1] : 0.0f;
    #pragma unroll
    for (int v = 0; v < 8; ++v) {
        C[(size_t)(rbase + v) * 256 + col0] = acc0[v] + bv0;
        C[(size_t)(rbase + v) * 256 + col1] = acc1[v] + bv1;
    }
}

// --- degree counting: inv[n][r] = 1 / max(#incoming edges of relation r, 1) ---
__global__ void zero_f32_kernel(float* __restrict__ p, int n) {
    int i = blockIdx.x * blockDim.x + threadIdx.x;
    if (i < n) p[i] = 0.0f;
}

__global__ void count_kernel(const int* __restrict__ dst, const int* __restrict__ et,
                             float* __restrict__ cnt, int E) {
    int e = blockIdx.x * blockDim.x + threadIdx.x;
    if (e < E) atomicAdd(&cnt[dst[e] * 3 + et[e]], 1.0f);
}

__global__ void invert_kernel(float* __restrict__ cnt, int n) {
    int i = blockIdx.x * blockDim.x + threadIdx.x;
    if (i < n) cnt[i] = 1.0f / fmaxf(cnt[i], 1.0f);
}

// --- edge scatter: accum[dst][c] += h_rel[src][c] * inv_deg[dst][rel] ---
// Reads cols 64..255 (relation transforms), writes cols 0..63 (base) of the
// SAME h_all buffer: disjoint column ranges -> no races.
__global__ __launch_bounds__(256)
void rgcn_scatter(float* __restrict__ hall,          // [N][256]
                  const float* __restrict__ inv,     // [N][3]
                  const int* __restrict__ src,
                  const int* __restrict__ dst,
                  const int* __restrict__ et, int E)
{
    int e = blockIdx.x * 4 + (threadIdx.x >> 6);  // 4 edges / block, 64 threads each
    if (e >= E) return;
    int c = threadIdx.x & 63;
    int s = src[e], d = dst[e], r = et[e];
    float w = inv[d * 3 + r];
    float v = hall[(size_t)s * 256 + 64 + r * 64 + c] * w;
    atomicAdd(&hall[(size_t)d * 256 + c], v);
}

// --- final classifier: out[n][c] = bc[c] + sum_k relu(h2[n][k]) * Wc[k][c] ---
__global__ void classifier_kernel(const float* __restrict__ hall,  // [N][256], base cols
                                  const float* __restrict__ Wc,    // [64][5]
                                  const float* __restrict__ bc,    // [5]
                                  float* __restrict__ out, int n)
{
    int idx  = blockIdx.x * blockDim.x + threadIdx.x;
    int node = idx >> 3;
    int c    = idx & 7;
    if (node >= n || c >= 5) return;
    const float* h = hall + (size_t)node * 256;
    float acc = bc[c];
    #pragma unroll
    for (int k = 0; k < 64; ++k)
        acc += fmaxf(h[k], 0.0f) * Wc[k * 5 + c];
    out[node * 5 + c] = acc;
}

extern "C" void kernel_launch(void* const* d_in, const int* in_sizes, int n_in,
                              void* d_out, int out_size, void* d_ws, size_t ws_size,
                              hipStream_t stream) {
    (void)n_in; (void)out_size; (void)ws_size;
    const float* x     = (const float*)d_in[0];
    const float* W1    = (const float*)d_in[1];
    const float* root1 = (const float*)d_in[2];
    const float* b1    = (const float*)d_in[3];
    const float* W2    = (const float*)d_in[4];
    const float* root2 = (const float*)d_in[5];
    const float* b2    = (const float*)d_in[6];
    const float* Wc    = (const float*)d_in[7];
    const float* bc    = (const float*)d_in[8];
    const int*   eidx  = (const int*)d_in[9];
    const int*   et    = (const int*)d_in[10];

    const int N = in_sizes[0] / 64;   // 50000 (divisible by 16)
    const int E = in_sizes[10];       // 800000
    const int* src = eidx;            // edge_index row 0
    const int* dst = eidx + E;        // edge_index row 1

    float* hall = (float*)d_ws;                 // [N][256]
    float* cnt  = hall + (size_t)N * 256;       // [N][3] -> inverse degrees
    float* out  = (float*)d_out;

    const int NR = N * 3;

    // Inverse per-relation in-degrees (shared by both layers)
    zero_f32_kernel<<<(NR + 255) / 256, 256, 0, stream>>>(cnt, NR);
    count_kernel   <<<(E  + 255) / 256, 256, 0, stream>>>(dst, et, cnt, E);
    invert_kernel  <<<(NR + 255) / 256, 256, 0, stream>>>(cnt, NR);

    // Layer 1: h_all = [x@root1+b1 | x@W1[0] | x@W1[1] | x@W1[2]], then scatter means
    rgcn_gemm_wmma<<<N / 16, 256, 0, stream>>>(x, 64, 0, root1, W1, b1, hall);
    rgcn_scatter  <<<E / 4, 256, 0, stream>>>(hall, cnt, src, dst, et, E);

    // Layer 2 (in-place): A = relu(base cols of h_all), rewritten as new h_all
    rgcn_gemm_wmma<<<N / 16, 256, 0, stream>>>(hall, 256, 1, root2, W2, b2, hall);
    rgcn_scatter  <<<E / 4, 256, 0, stream>>>(hall, cnt, src, dst, et, E);

    // Classifier with fused final ReLU
    classifier_kernel<<<(N * 8 + 255) / 256, 256, 0, stream>>>(hall, Wc, bc, out, N);
}